// Model4po_32358283608161
// MI455X (gfx1250) — compile-verified
//
#include <hip/hip_runtime.h>

typedef __attribute__((ext_vector_type(2))) float v2f;
typedef __attribute__((ext_vector_type(8))) float v8f;

#define B_  64
#define S_  512
#define H_  768
#define P2_ 46

// ---------------------------------------------------------------------------
// Prep: per-batch cue length (mask reduction) and subject vector s = h[i0]+h[i1]
// ---------------------------------------------------------------------------
__global__ void __launch_bounds__(256) prep_kernel(
    const float* __restrict__ h,
    const int*   __restrict__ ids,
    const int*   __restrict__ mask,
    float*       __restrict__ sbuf,
    int*         __restrict__ cue) {
  const int b = blockIdx.x;
  const int t = threadIdx.x;

  __shared__ int red[256];
  int acc = 0;
  for (int i = t; i < S_; i += 256) acc += mask[b * S_ + i];
  red[t] = acc;
  __syncthreads();
  for (int off = 128; off > 0; off >>= 1) {
    if (t < off) red[t] += red[t + off];
    __syncthreads();
  }
  if (t == 0) cue[b] = red[0];

  const int i0 = ids[b * 2 + 0];
  const int i1 = ids[b * 2 + 1];
  const float* r0 = h + ((size_t)b * S_ + i0) * H_;
  const float* r1 = h + ((size_t)b * S_ + i1) * H_;
  float* sb = sbuf + (size_t)b * H_;
  for (int k = t; k < H_; k += 256) sb[k] = r0[k] + r1[k];
}

// ---------------------------------------------------------------------------
// Fused masked-add + GEMM (f32 WMMA 16x16x4) + bias + sigmoid^4
// One wave32 per 16-row sequence tile; 4 waves per block.
// ---------------------------------------------------------------------------
__global__ void __launch_bounds__(128) gemm_kernel(
    const float* __restrict__ h,
    const float* __restrict__ W,
    const float* __restrict__ bias,
    const float* __restrict__ sbuf,
    const int*   __restrict__ cue,
    float*       __restrict__ out) {
  const int lane   = threadIdx.x & 31;
  const int waveId = threadIdx.x >> 5;
  const int tile   = blockIdx.x * 4 + waveId;   // 0 .. B_*S_/16 - 1

  const int b    = tile >> 5;                   // 32 row-tiles per batch
  const int s0   = (tile & 31) << 4;            // first seq row of tile
  const int half = lane >> 4;                   // 0: K lo pair, 1: K hi pair
  const int lm   = lane & 15;                   // A: row m ; B: col n

  const int   cueLen = cue[b];
  const float smask  = ((s0 + lm) < cueLen) ? 1.0f : 0.0f;

  // A-fragment source: row (s0+lm) of hidden, starting at K offset 2*half
  const float* hrow = h    + ((size_t)b * S_ + (size_t)(s0 + lm)) * H_ + half * 2;
  const float* srow = sbuf + (size_t)b * H_ + half * 2;

  // B-fragment sources: W rows p = 16*t + lm (clamped; padded cols never stored)
  const int p0 = lm;
  const int p1 = 16 + lm;
  int       p2 = 32 + lm; if (p2 >= P2_) p2 = P2_ - 1;
  const float* wrow0 = W + (size_t)p0 * H_ + half * 2;
  const float* wrow1 = W + (size_t)p1 * H_ + half * 2;
  const float* wrow2 = W + (size_t)p2 * H_ + half * 2;

  v8f acc0 = {}, acc1 = {}, acc2 = {};

#pragma unroll 4
  for (int k = 0; k < H_; k += 4) {
    v2f a  = *(const v2f*)(hrow + k);
    v2f sv = *(const v2f*)(srow + k);
    a.x = __builtin_fmaf(smask, sv.x, a.x);
    a.y = __builtin_fmaf(smask, sv.y, a.y);

    v2f b0 = *(const v2f*)(wrow0 + k);
    v2f b1 = *(const v2f*)(wrow1 + k);
    v2f b2 = *(const v2f*)(wrow2 + k);

    acc0 = __builtin_amdgcn_wmma_f32_16x16x4_f32(false, a, false, b0,
                                                 (short)0, acc0, false, false);
    acc1 = __builtin_amdgcn_wmma_f32_16x16x4_f32(false, a, false, b1,
                                                 (short)0, acc1, false, false);
    acc2 = __builtin_amdgcn_wmma_f32_16x16x4_f32(false, a, false, b2,
                                                 (short)0, acc2, false, false);
  }

  // Epilogue: D layout -> (m = r + 8*half, n = lm), p = 16*t + n
  const size_t rowBase = ((size_t)b * S_ + s0) * P2_;

#pragma unroll
  for (int t = 0; t < 3; ++t) {
    const int p = 16 * t + lm;
    if (p >= P2_) continue;                     // after all WMMAs: divergence OK
    const float bb = bias[p];
    const v8f acc = (t == 0) ? acc0 : (t == 1) ? acc1 : acc2;
#pragma unroll
    for (int r = 0; r < 8; ++r) {
      const int m = r + 8 * half;
      float v  = acc[r] + bb;
      // sigmoid(v)^4 with raw v_rcp_f32 (sufficient precision for sigmoid^4)
      float sg = __builtin_amdgcn_rcpf(1.0f + __expf(-v));
      float s2 = sg * sg;
      out[rowBase + (size_t)m * P2_ + p] = s2 * s2;
    }
  }
}

// ---------------------------------------------------------------------------
// Host launch
// ---------------------------------------------------------------------------
extern "C" void kernel_launch(void* const* d_in, const int* in_sizes, int n_in,
                              void* d_out, int out_size, void* d_ws, size_t ws_size,
                              hipStream_t stream) {
  const float* h    = (const float*)d_in[0];   // [B,S,H]
  const int*   ids  = (const int*)  d_in[1];   // [B,2]
  const int*   mask = (const int*)  d_in[2];   // [B,S]
  const float* W    = (const float*)d_in[3];   // [P2,H]
  const float* bias = (const float*)d_in[4];   // [P2]
  float*       out  = (float*)d_out;           // [B,S,P2]

  int*   cue  = (int*)d_ws;                          // 64 ints
  float* sbuf = (float*)((char*)d_ws + 256);         // [B,H] floats

  prep_kernel<<<B_, 256, 0, stream>>>(h, ids, mask, sbuf, cue);

  const int nTiles = (B_ * S_) / 16;                 // 2048 waves
  gemm_kernel<<<nTiles / 4, 128, 0, stream>>>(h, W, bias, sbuf, cue, out);
}